// SE2_55224689492532
// MI455X (gfx1250) — compile-verified
//
#include <hip/hip_runtime.h>

typedef __attribute__((ext_vector_type(2))) float v2f;
typedef __attribute__((ext_vector_type(4))) float f4;
typedef __attribute__((ext_vector_type(8))) float v8f;

// Problem constants (reference: B=16, C=256, H=128, W=128, sq=256)
#define Bn 16
#define Cn 256
#define Hn 128
#define Wn 128
#define ZPITCH 260  // LDS row pitch for z (16x256): 260 % 64banks = 4 -> conflict-free

// ---------------------------------------------------------------------------
// Kernel 1: quadrant means.  blockIdx.x = b*1024 + q*256 + c  (== means layout)
// 256 threads, each loads 4x float4 (16 floats) of the 64x64 quadrant.
// ---------------------------------------------------------------------------
__global__ void __launch_bounds__(256)
quad_mean_kernel(const float* __restrict__ t, float* __restrict__ means) {
    int blk = blockIdx.x;
    int b = blk >> 10;
    int q = (blk >> 8) & 3;
    int c = blk & 255;
    const float* base = t + ((size_t)(b * Cn + c) << 14)   // * H*W
                          + (size_t)((q >> 1) << 6) * Wn   // h0 * W
                          + ((q & 1) << 6);                // w0
    int tid = threadIdx.x;
    float s = 0.f;
#pragma unroll
    for (int i = 0; i < 4; ++i) {
        int lin = tid + (i << 8);          // 0..1023 float4 slots (64 rows x 16)
        int r   = lin >> 4;
        int c4  = lin & 15;
        f4 v = *(const f4*)(base + (size_t)r * Wn + (c4 << 2));
        s += v.x + v.y + v.z + v.w;
    }
    // wave32 butterfly reduce
#pragma unroll
    for (int off = 16; off > 0; off >>= 1)
        s += __shfl_xor(s, off, 32);
    __shared__ float red[8];
    int lane = tid & 31, wv = tid >> 5;
    if (lane == 0) red[wv] = s;
    __syncthreads();
    if (tid == 0) {
        float tot = 0.f;
#pragma unroll
        for (int i = 0; i < 8; ++i) tot += red[i];
        means[blk] = tot * (1.0f / 4096.0f);
    }
}

// ---------------------------------------------------------------------------
// Kernel 2: gates via V_WMMA_F32_16X16X4_F32 (fp32 WMMA, M=16 == batch).
// 1 block, 512 threads = 16 waves.  GEMM1: z(16,256)=means(16,1024)@Wr^T,
// mish, stage z in LDS; GEMM2: gate(16,1024)=z@We^T, sigmoid.
// A-frag: lane=M, vgpr j -> K = k0 + (lane>>4)*2 + j  (contiguous float2)
// B-frag: lane=N, same K mapping; W[n][k] row-major == W^T access pattern.
// C/D:    vgpr r -> row r + 8*(lane>>4), col = lane&15.
// ---------------------------------------------------------------------------
__device__ __forceinline__ float mish_f(float x) {
    float sp = (x > 20.f) ? x : log1pf(__expf(x));
    return x * tanhf(sp);
}
__device__ __forceinline__ float sigmoid_f(float x) {
    return 1.0f / (1.0f + __expf(-x));
}

__global__ void __launch_bounds__(512)
se_gate_kernel(const float* __restrict__ means,
               const float* __restrict__ w_reduce, const float* __restrict__ b_reduce,
               const float* __restrict__ w_expand, const float* __restrict__ b_expand,
               float* __restrict__ gates) {
    __shared__ float zs[16 * ZPITCH];
    int lane = threadIdx.x & 31;
    int wv   = threadIdx.x >> 5;   // 0..15
    int nlo  = lane & 15;          // M for A-frag / N for B-frag / col for C
    int khi  = lane >> 4;          // K half-select; also row-half for C

    // ---- GEMM1: wave wv -> z columns [wv*16, wv*16+16)
    {
        int c0 = wv << 4;
        v8f acc = {};
        const float* ap = means    + (size_t)nlo * 1024        + (khi << 1);
        const float* bp = w_reduce + (size_t)(c0 + nlo) * 1024 + (khi << 1);
        for (int k0 = 0; k0 < 1024; k0 += 4) {
            v2f a = *(const v2f*)(ap + k0);
            v2f b = *(const v2f*)(bp + k0);
            acc = __builtin_amdgcn_wmma_f32_16x16x4_f32(
                false, a, false, b, (short)0, acc, false, false);
        }
        float bias = b_reduce[c0 + nlo];
#pragma unroll
        for (int r = 0; r < 8; ++r) {
            int m = r + (khi << 3);
            zs[m * ZPITCH + c0 + nlo] = mish_f(acc[r] + bias);
        }
    }
    __syncthreads();

    // ---- GEMM2: wave wv -> 4 tiles of 16 gate columns
#pragma unroll
    for (int tile = 0; tile < 4; ++tile) {
        int c0 = ((wv << 2) + tile) << 4;
        v8f acc = {};
        const float* ap = zs       + nlo * ZPITCH             + (khi << 1);
        const float* bp = w_expand + (size_t)(c0 + nlo) * 256 + (khi << 1);
        for (int k0 = 0; k0 < 256; k0 += 4) {
            v2f a = *(const v2f*)(ap + k0);
            v2f b = *(const v2f*)(bp + k0);
            acc = __builtin_amdgcn_wmma_f32_16x16x4_f32(
                false, a, false, b, (short)0, acc, false, false);
        }
        float bias = b_expand[c0 + nlo];
#pragma unroll
        for (int r = 0; r < 8; ++r) {
            int m = r + (khi << 3);
            gates[m * 1024 + c0 + nlo] = sigmoid_f(acc[r] + bias);
        }
    }
}

// ---------------------------------------------------------------------------
// Kernel 3: out = t * gate[b, q*C + c], q = (h>=64)*2 + (w>=64).
// float4 per thread (never crosses the w=64 boundary).  Non-temporal store so
// the 256 MB output stream doesn't evict the L2-resident input.
// ---------------------------------------------------------------------------
__global__ void __launch_bounds__(256)
apply_gate_kernel(const float* __restrict__ t,
                  const float* __restrict__ gates,
                  float* __restrict__ out) {
    int i = blockIdx.x * 256 + threadIdx.x;   // float4 index: 0 .. 2^24-1
    f4 v = *(const f4*)(t + ((size_t)i << 2));
    int w4 = i & 31;            // float4 column within the 128-wide row
    int h  = (i >> 5) & 127;
    int c  = (i >> 12) & 255;
    int b  = i >> 20;
    int q  = ((h >> 6) << 1) | (w4 >> 4);
    float g = gates[(b << 10) + (q << 8) + c];
    f4 o = v * g;
    __builtin_nontemporal_store(o, (f4*)(out + ((size_t)i << 2)));
}

// ---------------------------------------------------------------------------
extern "C" void kernel_launch(void* const* d_in, const int* in_sizes, int n_in,
                              void* d_out, int out_size, void* d_ws, size_t ws_size,
                              hipStream_t stream) {
    const float* t        = (const float*)d_in[0];
    const float* w_reduce = (const float*)d_in[1];
    const float* b_reduce = (const float*)d_in[2];
    const float* w_expand = (const float*)d_in[3];
    const float* b_expand = (const float*)d_in[4];
    float* out = (float*)d_out;

    float* means = (float*)d_ws;          // 16*1024 floats
    float* gates = means + Bn * 4 * Cn;   // 16*1024 floats

    // Pass 1: 16384 (b,q,c) reductions over 64x64 quadrants
    quad_mean_kernel<<<Bn * 4 * Cn, 256, 0, stream>>>(t, means);

    // Pass 2: tiny FC stack on one WGP via fp32 WMMA
    se_gate_kernel<<<1, 512, 0, stream>>>(means, w_reduce, b_reduce,
                                          w_expand, b_expand, gates);

    // Pass 3: 16M float4 scale + reassemble
    apply_gate_kernel<<<(Bn * Cn * Hn * Wn / 4) / 256, 256, 0, stream>>>(t, gates, out);
}